// TNTLocalBranch_2362232012942
// MI455X (gfx1250) — compile-verified
//
#include <hip/hip_runtime.h>

typedef __attribute__((ext_vector_type(2))) float v2f;
typedef __attribute__((ext_vector_type(8))) float v8f;

#define DD 64
#define SS 64
#define HH 8
#define LRc 0.01f
#define WDc 0.01f

// Fast tanh: prefer the CDNA5 hardware TRANS op, fall back to 2-TRANS form.
__device__ __forceinline__ float fast_tanh(float x) {
#if __has_builtin(__builtin_amdgcn_tanhf)
    return __builtin_amdgcn_tanhf(x);
#elif __has_builtin(__builtin_amdgcn_tanh_f32)
    return __builtin_amdgcn_tanh_f32(x);
#else
    // tanh(x) = 1 - 2/(1 + e^{2x});  e^{2x} = exp2(2*log2(e)*x)
    // exp2 overflow -> +inf -> rcp=0 -> +1 ; underflow -> 0 -> -1  (correct saturation)
    float e = __builtin_amdgcn_exp2f(2.8853900817779268f * x);
    return 1.0f - 2.0f * __builtin_amdgcn_rcpf(e + 1.0f);
#endif
}

// One workgroup per shard. 512 threads = 16 waves (wave32).
// LDS: Xs (16KB) + E (16KB) + Q (16KB) + double-buffered reduce (4KB) = 52KB.
__global__ __launch_bounds__(512) void tnt_fused_kernel(
    const float* __restrict__ X,
    const float* __restrict__ init_mem,
    const float* __restrict__ init_opt,
    const float* __restrict__ Wq,
    float* __restrict__ out)
{
    __shared__ float sXs[SS * DD];
    __shared__ float sE [SS * DD];
    __shared__ float sQ [SS * DD];
    __shared__ float sRed[2][512];

    const int tid = threadIdx.x;
    const int n   = blockIdx.x;
    const float* __restrict__ xs = X + (size_t)n * SS * DD;

    // Stage this shard's X tile into LDS (b128 path).
    {
        const float4* __restrict__ xg = (const float4*)xs;
        float4* sx = (float4*)sXs;
        #pragma unroll
        for (int i = 0; i < (SS * DD / 4); i += 512) sx[i + tid] = xg[i + tid];
    }
    __syncthreads();

    // ---- Front GEMMs with V_WMMA_F32_16X16X4_F32 (fp32-exact) ----
    // 16 waves <-> 16 tiles (4x4 of 16x16); each wave computes its E-tile and
    // Q-tile together, sharing the A fragment (no divergent selection).
    const int wave = tid >> 5;
    const int lane = tid & 31;
    const int lh   = lane & 15;            // lane-half index
    const int khi  = (lane >> 4) << 1;     // K offset 0 or 2 (32-bit A/B layout)
    const int rofs = (lane >> 4) << 3;     // C/D row offset 0 or 8

    {
        const int m0 = (wave >> 2) << 4;
        const int n0 = (wave & 3) << 4;
        v8f accE = {};
        v8f accQ = {};
        #pragma unroll
        for (int k0 = 0; k0 < DD; k0 += 4) {
            v2f a, bW, bQ;
            // A 16x4 f32: lanes 0-15 -> K = k0+{0,1}; lanes 16-31 -> K = k0+{2,3}
            a.x = sXs[(m0 + lh) * DD + k0 + khi + 0];
            a.y = sXs[(m0 + lh) * DD + k0 + khi + 1];
            // B 4x16 f32: same K split, N striped across lanes (L2-hot 16KB each)
            bW.x = init_mem[(k0 + khi + 0) * DD + n0 + lh];
            bW.y = init_mem[(k0 + khi + 1) * DD + n0 + lh];
            bQ.x = Wq[(k0 + khi + 0) * DD + n0 + lh];
            bQ.y = Wq[(k0 + khi + 1) * DD + n0 + lh];
            accE = __builtin_amdgcn_wmma_f32_16x16x4_f32(
                false, a, false, bW, (short)0, accE, false, false);
            accQ = __builtin_amdgcn_wmma_f32_16x16x4_f32(
                false, a, false, bQ, (short)0, accQ, false, false);
        }
        #pragma unroll
        for (int r = 0; r < 8; ++r) {
            const int row = m0 + r + rofs;
            const int col = n0 + lh;
            sE[row * DD + col] = accE[r] - sXs[row * DD + col];  // pred - Xs
            sQ[row * DD + col] = accQ[r];                        // Xs @ Wq
        }
    }
    __syncthreads();

    // ---- Fused MLP + cumsum + matvec; memory matrix lives in registers ----
    const int k  = tid & 63;        // output column owned by this thread
    const int dg = tid >> 6;        // 0..7 -> 8 d-rows per thread
    const int d0 = dg << 3;

    float w1r[HH], b1r[HH], w2r[HH];
    #pragma unroll
    for (int j = 0; j < HH; ++j) {
        w1r[j] = init_opt[j];
        b1r[j] = init_opt[HH + j];
        w2r[j] = init_opt[2 * HH + j];
    }
    const float b2 = init_opt[3 * HH];

    float M[8], W0r[8];
    #pragma unroll
    for (int i = 0; i < 8; ++i) {
        W0r[i] = init_mem[(d0 + i) * DD + k];   // coalesced, L2-hot
        M[i]   = W0r[i];
    }

    const float cg = 2.0f / (float)DD;
    float* __restrict__ outp = out + (size_t)n * SS * DD;

    for (int s = 0; s < SS; ++s) {
        const float  ek  = sE[s * DD + k];
        const float4 x4a = *(const float4*)(sXs + s * DD + d0);
        const float4 x4b = *(const float4*)(sXs + s * DD + d0 + 4);
        const float4 q4a = *(const float4*)(sQ  + s * DD + d0);
        const float4 q4b = *(const float4*)(sQ  + s * DD + d0 + 4);
        const float xv[8] = {x4a.x, x4a.y, x4a.z, x4a.w, x4b.x, x4b.y, x4b.z, x4b.w};
        const float qv[8] = {q4a.x, q4a.y, q4a.z, q4a.w, q4b.x, q4b.y, q4b.z, q4b.w};

        float partial = 0.0f;
        #pragma unroll
        for (int i = 0; i < 8; ++i) {
            const float g = cg * xv[i] * ek;
            float base = b2;
            #pragma unroll
            for (int j = 0; j < HH; ++j)
                base = __builtin_fmaf(fast_tanh(__builtin_fmaf(g, w1r[j], b1r[j])),
                                      w2r[j], base);
            M[i] -= LRc * base + WDc * W0r[i];          // inclusive cumsum state
            partial = __builtin_fmaf(qv[i], M[i], partial);
        }

        // Double-buffered cross-wave reduction: one barrier per step.
        float* red = sRed[s & 1];
        red[tid] = partial;
        __syncthreads();
        if (tid < 64) {
            float o = 0.0f;
            #pragma unroll
            for (int r = 0; r < 8; ++r) o += red[r * 64 + k];
            outp[s * DD + k] = o;   // coalesced 256B row per step
        }
    }
}

extern "C" void kernel_launch(void* const* d_in, const int* in_sizes, int n_in,
                              void* d_out, int out_size, void* d_ws, size_t ws_size,
                              hipStream_t stream) {
    const float* X        = (const float*)d_in[0];
    const float* init_mem = (const float*)d_in[1];
    const float* init_opt = (const float*)d_in[2];
    const float* Wq       = (const float*)d_in[3];
    float* out = (float*)d_out;

    const int nB = in_sizes[0] / (SS * DD);   // 128 shards
    tnt_fused_kernel<<<dim3(nB), dim3(512), 0, stream>>>(X, init_mem, init_opt, Wq, out);
}